// SO3Convolution_76957224010212
// MI455X (gfx1250) — compile-verified
//
#include <hip/hip_runtime.h>
#include <hip/hip_bf16.h>

typedef __attribute__((ext_vector_type(2))) float v2f;
typedef __attribute__((ext_vector_type(8))) float v8f;

#define FDIM 9      // (LMAX+1)^2
#define NF   128    // N_BASIS

// ---------------------------------------------------------------------------
// Kernel 0: zero the output accumulator (harness poisons d_out with 0xAA).
// ---------------------------------------------------------------------------
__global__ void so3_zero_kernel(float* __restrict__ out, int n) {
    int i = blockIdx.x * blockDim.x + threadIdx.x;
    if (i < n) out[i] = 0.0f;
}

// ---------------------------------------------------------------------------
// Wij = (radial[E,K] @ Wf[K,N] + b[N]) * cutoff[E]   via f32 WMMA.
// One 16x16 output tile per wave32; V_WMMA_F32_16X16X4_F32, K in steps of 4.
//
// Fragment layouts (ISA 7.12.2, 32-bit, wave32):
//   A (16x4, MxK): lanes 0-15: M=lane,   K={k0,k0+1};
//                  lanes16-31: M=lane-16,K={k0+2,k0+3}
//   B (4x16, KxN): mirrored (N=lane&15, K split by lane half)
//   C/D (16x16):   VGPR r: lanes 0-15 -> M=r; lanes 16-31 -> M=r+8; N=lane&15
//
// Out-of-range A rows are CLAMPED (not predicated): output row m depends only
// on A row m, and rows >= E are masked at the store. This keeps EXEC all-ones
// through the whole WMMA sequence and removes all branches from the K-loop.
// ---------------------------------------------------------------------------

// Fully-unrolled compile-time-K variant (K must be a multiple of 4).
template <int KT>
__global__ __launch_bounds__(128) void so3_wij_wmma_fixedK_kernel(
        const float* __restrict__ radial,   // [E, KT]
        const float* __restrict__ Wf,       // [KT, N]
        const float* __restrict__ bf,       // [N]
        const float* __restrict__ cutoff,   // [E, 1]
        float* __restrict__ Wij,            // [E, N]  (workspace)
        int E, int N) {
    constexpr int KS = KT / 4;              // # of WMMA k-steps
    const int wavesPerBlock = blockDim.x >> 5;
    const int wave   = blockIdx.x * wavesPerBlock + (threadIdx.x >> 5);
    const int lane   = threadIdx.x & 31;
    const int tilesN = N >> 4;
    const int tilesE = (E + 15) >> 4;
    const int tn = wave % tilesN;
    const int te = wave / tilesN;
    if (te >= tilesE) return;

    const int half = lane >> 4;             // lane half selects K pair
    const int mn   = lane & 15;

    const int arow  = te * 16 + mn;
    const int arowc = arow < E ? arow : (E - 1);   // clamp, no predication
    const int bcol  = tn * 16 + mn;

    // ---- load all fragments up front (A pairs contiguous -> b64 loads) ----
    v2f a[KS], b[KS];
    const float* ap = radial + (size_t)arowc * KT + 2 * half;
#pragma unroll
    for (int i = 0; i < KS; ++i) {
        a[i].x = ap[4 * i + 0];
        a[i].y = ap[4 * i + 1];
    }
#pragma unroll
    for (int i = 0; i < KS; ++i) {
        const int kb = 4 * i + 2 * half;
        b[i].x = Wf[(size_t)(kb + 0) * N + bcol];
        b[i].y = Wf[(size_t)(kb + 1) * N + bcol];
    }

    // ---- back-to-back WMMA chain ----
    v8f acc = {};
#pragma unroll
    for (int i = 0; i < KS; ++i) {
        // (neg_a, A, neg_b, B, c_mod, C, reuse_a, reuse_b)
        acc = __builtin_amdgcn_wmma_f32_16x16x4_f32(
                false, a[i], false, b[i], (short)0, acc, false, false);
    }

    // ---- epilogue: bias + cutoff, masked store ----
    const float bias = bf[bcol];
#pragma unroll
    for (int r = 0; r < 8; ++r) {
        const int row = te * 16 + r + 8 * half;
        if (row < E) {
            Wij[(size_t)row * N + bcol] = (acc[r] + bias) * cutoff[row];
        }
    }
}

// Generic runtime-K fallback: unguarded main loop over K&~3, guarded tail.
__global__ __launch_bounds__(128) void so3_wij_wmma_kernel(
        const float* __restrict__ radial, const float* __restrict__ Wf,
        const float* __restrict__ bf, const float* __restrict__ cutoff,
        float* __restrict__ Wij, int E, int K, int N) {
    const int wavesPerBlock = blockDim.x >> 5;
    const int wave   = blockIdx.x * wavesPerBlock + (threadIdx.x >> 5);
    const int lane   = threadIdx.x & 31;
    const int tilesN = N >> 4;
    const int tilesE = (E + 15) >> 4;
    const int tn = wave % tilesN;
    const int te = wave / tilesN;
    if (te >= tilesE) return;

    const int half = lane >> 4;
    const int mn   = lane & 15;
    const int arow  = te * 16 + mn;
    const int arowc = arow < E ? arow : (E - 1);
    const int bcol  = tn * 16 + mn;
    const float* ap = radial + (size_t)arowc * K;

    v8f acc = {};
    const int Kmain = K & ~3;
    for (int k0 = 0; k0 < Kmain; k0 += 4) {
        const int ka = k0 + 2 * half;
        v2f a, b;
        a.x = ap[ka + 0];
        a.y = ap[ka + 1];
        b.x = Wf[(size_t)(ka + 0) * N + bcol];
        b.y = Wf[(size_t)(ka + 1) * N + bcol];
        acc = __builtin_amdgcn_wmma_f32_16x16x4_f32(
                false, a, false, b, (short)0, acc, false, false);
    }
    if (Kmain < K) {                       // zero-padded tail (K % 4 != 0)
        const int ka = Kmain + 2 * half;
        v2f a, b;
        a.x = (ka + 0 < K) ? ap[ka + 0] : 0.0f;
        a.y = (ka + 1 < K) ? ap[ka + 1] : 0.0f;
        b.x = (ka + 0 < K) ? Wf[(size_t)(ka + 0) * N + bcol] : 0.0f;
        b.y = (ka + 1 < K) ? Wf[(size_t)(ka + 1) * N + bcol] : 0.0f;
        acc = __builtin_amdgcn_wmma_f32_16x16x4_f32(
                false, a, false, b, (short)0, acc, false, false);
    }

    const float bias = bf[bcol];
#pragma unroll
    for (int r = 0; r < 8; ++r) {
        const int row = te * 16 + r + 8 * half;
        if (row < E) {
            Wij[(size_t)row * N + bcol] = (acc[r] + bias) * cutoff[row];
        }
    }
}

// ---------------------------------------------------------------------------
// Kernel 2: per-edge sparse CG tensor product + atomic scatter to atoms.
// One edge per block; one feature channel per thread. Dynamic (s1,s2,s3,l)
// indices resolved via LDS arrays in [s][f] layout: with uniform s across the
// wave, all ds accesses are stride-1 / conflict-free.
// ---------------------------------------------------------------------------
__global__ __launch_bounds__(NF) void so3_cg_scatter_kernel(
        const float* __restrict__ x,        // [A, 9, NF]
        const float* __restrict__ dir,      // [E, 3]
        const float* __restrict__ Wij,      // [E, 3*NF]
        const float* __restrict__ cg_vals,  // [ncg]
        const int*   __restrict__ idx_i,    // [E]
        const int*   __restrict__ idx_j,    // [E]
        const int*   __restrict__ cg_i1,    // [ncg]
        const int*   __restrict__ cg_i2,    // [ncg]
        const int*   __restrict__ cg_i3,    // [ncg]
        const int*   __restrict__ cg_wl,    // [ncg] (= l of output index)
        float* __restrict__ out,            // [A, 9, NF]
        int n_cg, int E) {
    const int e = blockIdx.x;
    if (e >= E) return;
    const int f = threadIdx.x;

    __shared__ float Ysh[FDIM];
    __shared__ float xl [FDIM * NF];   // neighbor features   [s][f]
    __shared__ float wls[3 * NF];      // per-l radial filter [l][f]
    __shared__ float accl[FDIM * NF];  // output accumulator  [s][f]

    const int j = idx_j[e];
    const float* xrow = x + ((size_t)j * FDIM) * NF + f;
    __builtin_prefetch(xrow, 0, 1);    // global_prefetch_b8 on gathered row

    // --- spherical harmonics (one thread, broadcast via LDS) ---
    if (f == 0) {
        float dx = dir[e * 3 + 0];
        float dy = dir[e * 3 + 1];
        float dz = dir[e * 3 + 2];
        float inv = rsqrtf(dx * dx + dy * dy + dz * dz);
        float X = dx * inv, Y = dy * inv, Z = dz * inv;
        const float s3  = 1.7320508075688772f;   // sqrt(3)
        const float s5  = 2.2360679774997896f;   // sqrt(5)
        const float s15 = 3.8729833462074170f;   // sqrt(15)
        Ysh[0] = 1.0f;
        Ysh[1] = s3 * Y;
        Ysh[2] = s3 * Z;
        Ysh[3] = s3 * X;
        Ysh[4] = s15 * X * Y;
        Ysh[5] = s15 * Y * Z;
        Ysh[6] = 0.5f * s5 * (3.0f * Z * Z - 1.0f);
        Ysh[7] = s15 * X * Z;
        Ysh[8] = 0.5f * s15 * (X * X - Y * Y);
    }

    // --- stage per-thread operands into LDS ---
#pragma unroll
    for (int s = 0; s < FDIM; ++s) xl[s * NF + f] = xrow[(size_t)s * NF];

    const float* wrow = Wij + (size_t)e * (3 * NF) + f;
#pragma unroll
    for (int l = 0; l < 3; ++l) wls[l * NF + f] = wrow[l * NF];

#pragma unroll
    for (int s = 0; s < FDIM; ++s) accl[s * NF + f] = 0.0f;

    __syncthreads();

    // --- sparse CG contraction (index lists broadcast from L2/WGP$) ---
    for (int n = 0; n < n_cg; ++n) {
        const float v  = cg_vals[n];
        const int   s1 = cg_i1[n];
        const int   s2 = cg_i2[n];
        const int   s3 = cg_i3[n];
        const int   l  = cg_wl[n];
        accl[s3 * NF + f] += Ysh[s2] * v * wls[l * NF + f] * xl[s1 * NF + f];
    }

    // --- scatter-add into receiving atom ---
    const int i = idx_i[e];
    float* orow = out + ((size_t)i * FDIM) * NF + f;
#pragma unroll
    for (int s = 0; s < FDIM; ++s)
        atomicAdd(&orow[(size_t)s * NF], accl[s * NF + f]);
}

// ---------------------------------------------------------------------------
// Host-side launcher. Input order (setup_inputs):
//  0 x [A,9,128] f32 | 1 radial [E,20] | 2 dir [E,3] | 3 cutoff [E,1]
//  4 W_filter [20,384] | 5 b_filter [384] | 6 cg_vals [ncg]
//  7 idx_i [E] i32 | 8 idx_j [E] i32 | 9..12 cg idx lists [ncg] i32
// ---------------------------------------------------------------------------
extern "C" void kernel_launch(void* const* d_in, const int* in_sizes, int n_in,
                              void* d_out, int out_size, void* d_ws, size_t ws_size,
                              hipStream_t stream) {
    const float* x       = (const float*)d_in[0];
    const float* radial  = (const float*)d_in[1];
    const float* dir     = (const float*)d_in[2];
    const float* cutoff  = (const float*)d_in[3];
    const float* Wf      = (const float*)d_in[4];
    const float* bf      = (const float*)d_in[5];
    const float* cg_vals = (const float*)d_in[6];
    const int*   idx_i   = (const int*)d_in[7];
    const int*   idx_j   = (const int*)d_in[8];
    const int*   cg_i1   = (const int*)d_in[9];
    const int*   cg_i2   = (const int*)d_in[10];
    const int*   cg_i3   = (const int*)d_in[11];
    const int*   cg_wl   = (const int*)d_in[12];

    const int E    = in_sizes[2] / 3;           // 12000
    const int K    = in_sizes[1] / E;           // 20
    const int N    = in_sizes[4] / K;           // 384 = 3*128
    const int n_cg = in_sizes[6];

    float* Wij = (float*)d_ws;                  // [E, N] scratch, ~18.4 MB
    float* out = (float*)d_out;

    // 0) zero output
    {
        const int threads = 256;
        const int blocks  = (out_size + threads - 1) / threads;
        so3_zero_kernel<<<blocks, threads, 0, stream>>>(out, out_size);
    }

    // 1) radial filter GEMM via WMMA (one 16x16 tile per wave32)
    {
        const int tilesE = (E + 15) / 16;
        const int tilesN = N / 16;
        const int tiles  = tilesE * tilesN;
        const int threads = 128;                       // 4 waves / block
        const int wavesPerBlock = threads / 32;
        const int blocks = (tiles + wavesPerBlock - 1) / wavesPerBlock;
        if (K == 20) {
            so3_wij_wmma_fixedK_kernel<20><<<blocks, threads, 0, stream>>>(
                radial, Wf, bf, cutoff, Wij, E, N);
        } else {
            so3_wij_wmma_kernel<<<blocks, threads, 0, stream>>>(
                radial, Wf, bf, cutoff, Wij, E, K, N);
        }
    }

    // 2) per-edge CG tensor product + atomic scatter
    {
        so3_cg_scatter_kernel<<<E, NF, 0, stream>>>(
            x, dir, Wij, cg_vals, idx_i, idx_j,
            cg_i1, cg_i2, cg_i3, cg_wl, out, n_cg, E);
    }
}